// CausalSelfAttention_13357348290661
// MI455X (gfx1250) — compile-verified
//
#include <hip/hip_runtime.h>
#include <hip/hip_bf16.h>

typedef __attribute__((ext_vector_type(2))) float v2f;
typedef __attribute__((ext_vector_type(8))) float v8f;
typedef __attribute__((ext_vector_type(4))) unsigned int v4u;
typedef __attribute__((ext_vector_type(4))) int v4i;
typedef __attribute__((ext_vector_type(8))) int v8i;

#define T_SEQ 2048
#define D_MOD 1024
#define NHEAD 16
#define HDIM 64
#define QKV_PART 4194304   // 4096*1024 floats per q/k/v part
#define ATT_SCALE 0.03125f // 1/sqrt(1024)

__device__ __forceinline__ v8f wmma_f32(v2f a, v2f b, v8f c) {
  // D = A(16x4 f32) x B(4x16 f32) + C(16x16 f32)
  return __builtin_amdgcn_wmma_f32_16x16x4_f32(
      /*neg_a=*/false, a, /*neg_b=*/false, b,
      /*c_mod=*/(short)0, c, /*reuse_a=*/false, /*reuse_b=*/false);
}

// ---------------------------------------------------------------------------
// TDM: async DMA of a 2D tile (4-byte elements) global -> LDS, with optional
// LDS row padding (pad codes per CDNA5 D# group1 encoding).
//   tile0 elems per row, tile1 rows; row stride in tensor = stride0 elems.
// ---------------------------------------------------------------------------
__device__ __forceinline__ void tdm_load_2d(unsigned int lds_byte_addr,
                                            const float* gptr,
                                            unsigned int tensor_d0,
                                            unsigned int tensor_d1,
                                            unsigned int stride0_elems,
                                            unsigned int tile0,
                                            unsigned int tile1,
                                            unsigned int pad_int_code,
                                            unsigned int pad_amt_code) {
  unsigned long long ga = (unsigned long long)(size_t)gptr;
  v4u g0;
  g0[0] = 1u;                                 // count=1 (valid user descriptor)
  g0[1] = lds_byte_addr;                      // lds_addr
  g0[2] = (unsigned int)ga;                   // global_addr[31:0]
  g0[3] = (unsigned int)((ga >> 32) & 0x01ffffffu) | (2u << 30); // addr[56:32] | type=2
  unsigned int w0 = (2u << 16)                              // data_size = 4B
                    | (1u << 20)                            // pad_enable
                    | (pad_int_code << 22) | (pad_amt_code << 25);
  v8i g1;
  g1[0] = (int)w0;                                          // wg_mask=0 (no cluster)
  g1[1] = (int)((tensor_d0 & 0xffffu) << 16);               // dim0[15:0] @ bits63:48
  g1[2] = (int)(((tensor_d0 >> 16) & 0xffffu) | ((tensor_d1 & 0xffffu) << 16));
  g1[3] = (int)(((tensor_d1 >> 16) & 0xffffu) | ((tile0 & 0xffffu) << 16));
  g1[4] = (int)(tile1 & 0xffffu);                           // tile_dim1; tile_dim2=0
  g1[5] = (int)stride0_elems;                               // dim0_stride[31:0]
  g1[6] = 0;                                                // stride hi bits
  g1[7] = 0;
  v4i gz = {0, 0, 0, 0};
#if defined(__clang_major__) && (__clang_major__ >= 23)
  v8i gz8 = {0, 0, 0, 0, 0, 0, 0, 0};
  __builtin_amdgcn_tensor_load_to_lds(g0, g1, gz, gz, gz8, 0);
#else
  __builtin_amdgcn_tensor_load_to_lds(g0, g1, gz, gz, 0);
#endif
}

__device__ __forceinline__ unsigned int lds_off(const void* p) {
  return (unsigned int)(size_t)p; // generic LDS ptr: low 32 bits = LDS offset
}

// ---------------------------------------------------------------------------
// GEMM: C[M,N] = X[M,1024] @ W[N,1024]^T
// Block: 256 threads (8 waves). Block tile 128x64, wave tile 16x64, Kc=16.
// Tiles streamed global->LDS by TDM, double-buffered (wave 0 is the issuer).
// LDS row stride 20 via TDM padding (interval code 3 = 16 dw, amount code 3 = 4 dw).
// mode 0: plain row-major store to Out; mode 1: scatter into qkv workspace.
// ---------------------------------------------------------------------------
__global__ __launch_bounds__(256) void gemm_xwt_kernel(
    const float* __restrict__ X, const float* __restrict__ W,
    float* __restrict__ Out, float* __restrict__ QKV,
    int N, int nTilesN, int mode) {
  __shared__ __align__(16) float Xs[2][128 * 20];
  __shared__ __align__(16) float Ws[2][64 * 20];

  const int tid = threadIdx.x;
  const int wave = tid >> 5;
  const int lane = tid & 31;
  const int ln = lane & 15;   // M/N index within 16
  const int kl = lane >> 4;   // k-half select

  const int mt = blockIdx.x / nTilesN;
  const int nt = blockIdx.x % nTilesN;
  const int m0 = mt * 128;
  const int n0 = nt * 64;

  const bool issuer = (wave == 0);
  const float* Xb = X + (size_t)m0 * 1024;
  const float* Wb = W + (size_t)n0 * 1024;

  if (issuer) { // prologue: chunk 0 -> buffer 0
    tdm_load_2d(lds_off(&Xs[0][0]), Xb, 1024, 128, 1024, 16, 128, 3, 3);
    tdm_load_2d(lds_off(&Ws[0][0]), Wb, 1024, 64, 1024, 16, 64, 3, 3);
  }

  v8f acc[4] = {};

  for (int kk0 = 0; kk0 < 1024; kk0 += 16) {
    const int buf = (kk0 >> 4) & 1;
    if (issuer) __builtin_amdgcn_s_wait_tensorcnt(0);
    __syncthreads(); // data ready for all; prev compute done before reissue
    if (issuer && (kk0 + 16 < 1024)) {
      tdm_load_2d(lds_off(&Xs[buf ^ 1][0]), Xb + kk0 + 16, 1024, 128, 1024, 16, 128, 3, 3);
      tdm_load_2d(lds_off(&Ws[buf ^ 1][0]), Wb + kk0 + 16, 1024, 64, 1024, 16, 64, 3, 3);
    }
    const int arow = wave * 16 + ln;
#pragma unroll
    for (int s = 0; s < 4; ++s) {
      const int kb = s * 4 + 2 * kl;
      v2f a = *(const v2f*)(&Xs[buf][arow * 20 + kb]);
#pragma unroll
      for (int sub = 0; sub < 4; ++sub) {
        v2f b = *(const v2f*)(&Ws[buf][(sub * 16 + ln) * 20 + kb]);
        acc[sub] = wmma_f32(a, b, acc[sub]);
      }
    }
  }

  // Epilogue: C layout -> vgpr j = row (j + 8*kl), col = sub*16 + ln
  const int mrow0 = m0 + wave * 16;
  if (mode == 0) {
#pragma unroll
    for (int sub = 0; sub < 4; ++sub) {
      const int col = n0 + sub * 16 + ln;
#pragma unroll
      for (int j = 0; j < 8; ++j) {
        const int m = mrow0 + j + 8 * kl;
        Out[(size_t)m * N + col] = acc[sub][j];
      }
    }
  } else {
#pragma unroll
    for (int sub = 0; sub < 4; ++sub) {
      const int e = n0 + sub * 16 + ln;
      const int part = e >> 10;
      const int d = e & 1023;
      const int h = d >> 6, hd = d & 63;
#pragma unroll
      for (int j = 0; j < 8; ++j) {
        const int m = mrow0 + j + 8 * kl;
        const int b = m >> 11, t = m & 2047;
        QKV[(size_t)part * QKV_PART +
            (size_t)(((b << 4) + h) * T_SEQ + t) * HDIM + hd] = acc[sub][j];
      }
    }
  }
}

// ---------------------------------------------------------------------------
// Flash attention: one block = one (b,h) x 128-query slab. 8 waves, each owns
// a 16-query tile. Online softmax over key tiles of 16, causal-masked.
// K tile (stride 68) and V tile (stride 72) streamed by TDM, double-buffered.
// Q/K/V layout: [(b*16+h)*2048 + t]*64 + hd. Z: [4096][1024].
// ---------------------------------------------------------------------------
__global__ __launch_bounds__(256) void attn_kernel(
    const float* __restrict__ Q, const float* __restrict__ K,
    const float* __restrict__ V, float* __restrict__ Z) {
  __shared__ __align__(16) float Ks[2][16 * 68];   // K tile row-major
  __shared__ __align__(16) float Vs[2][16 * 72];   // V tile row-major
  __shared__ __align__(16) float Ssc[8][16 * 20];  // per-wave S scratch

  const int tid = threadIdx.x;
  const int wave = tid >> 5;
  const int lane = tid & 31;
  const int ln = lane & 15;
  const int kl = lane >> 4;

  const int qb = blockIdx.x & 15;   // 128-query slab index
  const int bh = blockIdx.x >> 4;   // b*16 + h
  const float* Qh = Q + (size_t)bh * T_SEQ * HDIM;
  const float* Kh = K + (size_t)bh * T_SEQ * HDIM;
  const float* Vh = V + (size_t)bh * T_SEQ * HDIM;

  const int qbase = qb * 128 + wave * 16;
  const int qrow = qbase + ln;
  const bool issuer = (wave == 0);

  // Q fragments for 16 K-steps (HD=64, K=4 each): qf[s] = Q[m][s*4+2*kl+{0,1}]
  v2f qf[16];
#pragma unroll
  for (int s = 0; s < 16; ++s)
    qf[s] = *(const v2f*)(Qh + (size_t)qrow * HDIM + s * 4 + 2 * kl);

  v8f zacc[4] = {};
  float mrow = -INFINITY;
  float lrow = 0.0f;

  const int nkt = qb * 8 + 8;      // key tiles covering slab's causal range
  const int mymax = qbase + 15;

  if (issuer) { // prologue: key tile 0 -> buffer 0
    tdm_load_2d(lds_off(&Ks[0][0]), Kh, 64, 16, 64, 64, 16, 5, 3); // stride 68
    tdm_load_2d(lds_off(&Vs[0][0]), Vh, 64, 16, 64, 64, 16, 5, 7); // stride 72
  }

  for (int kt = 0; kt < nkt; ++kt) {
    const int kb0 = kt * 16;
    const int buf = kt & 1;
    if (issuer) __builtin_amdgcn_s_wait_tensorcnt(0);
    __syncthreads();
    if (issuer && (kt + 1 < nkt)) {
      const float* Kn = Kh + (size_t)(kb0 + 16) * HDIM;
      const float* Vn = Vh + (size_t)(kb0 + 16) * HDIM;
      tdm_load_2d(lds_off(&Ks[buf ^ 1][0]), Kn, 64, 16, 64, 64, 16, 5, 3);
      tdm_load_2d(lds_off(&Vs[buf ^ 1][0]), Vn, 64, 16, 64, 64, 16, 5, 7);
    }

    if (kb0 <= mymax) { // wave-uniform predicate: EXEC stays all-ones
      // S = Q K^T (16x16), K dim = 64 -> 16 WMMAs
      v8f s_acc = {};
#pragma unroll
      for (int s = 0; s < 16; ++s) {
        const int kb = s * 4 + 2 * kl;
        v2f bfr = *(const v2f*)(&Ks[buf][ln * 68 + kb]);
        s_acc = wmma_f32(qf[s], bfr, s_acc);
      }
      // Scale + causal mask, spill S to per-wave LDS scratch in C layout
      float* sw = &Ssc[wave][0];
#pragma unroll
      for (int j = 0; j < 8; ++j) {
        const int m = j + 8 * kl;
        const int kg = kb0 + ln;
        const float val = (kg <= qbase + m) ? s_acc[j] * ATT_SCALE : -INFINITY;
        sw[m * 20 + ln] = val;
      }
      asm volatile("s_wait_dscnt 0" ::: "memory");
      // Each lane reads its full softmax row r = ln (both halves redundant)
      float pv[16];
      {
        const float4 r0 = *(const float4*)(sw + ln * 20 + 0);
        const float4 r1 = *(const float4*)(sw + ln * 20 + 4);
        const float4 r2 = *(const float4*)(sw + ln * 20 + 8);
        const float4 r3 = *(const float4*)(sw + ln * 20 + 12);
        pv[0] = r0.x; pv[1] = r0.y; pv[2] = r0.z; pv[3] = r0.w;
        pv[4] = r1.x; pv[5] = r1.y; pv[6] = r1.z; pv[7] = r1.w;
        pv[8] = r2.x; pv[9] = r2.y; pv[10] = r2.z; pv[11] = r2.w;
        pv[12] = r3.x; pv[13] = r3.y; pv[14] = r3.z; pv[15] = r3.w;
      }
      float tm = pv[0];
#pragma unroll
      for (int j = 1; j < 16; ++j) tm = fmaxf(tm, pv[j]);
      const float mnew = fmaxf(mrow, tm);
      const float f = __expf(mrow - mnew); // exp(-inf)=0 on first tile
      float ts = 0.0f;
#pragma unroll
      for (int j = 0; j < 16; ++j) {
        pv[j] = __expf(pv[j] - mnew);
        ts += pv[j];
      }
      lrow = lrow * f + ts;
      mrow = mnew;
      // Rescale Z accumulator: vgpr j needs factor of row j (lo half) / j+8 (hi)
      float fj[8];
#pragma unroll
      for (int j = 0; j < 8; ++j)
        fj[j] = __shfl(f, (lane < 16) ? j : 24 + j, 32);
#pragma unroll
      for (int sub = 0; sub < 4; ++sub)
#pragma unroll
        for (int j = 0; j < 8; ++j) zacc[sub][j] *= fj[j];
      // Z += P V : A = P rows (per-lane), B = V row-major fragments {v[0],v[72]}
#pragma unroll
      for (int s = 0; s < 4; ++s) {
        v2f a;
        a.x = kl ? pv[s * 4 + 2] : pv[s * 4 + 0];
        a.y = kl ? pv[s * 4 + 3] : pv[s * 4 + 1];
        const int kb = s * 4 + 2 * kl;
#pragma unroll
        for (int sub = 0; sub < 4; ++sub) {
          const float* vb = &Vs[buf][kb * 72 + sub * 16 + ln];
          v2f bfr;
          bfr.x = vb[0];
          bfr.y = vb[72];
          zacc[sub] = wmma_f32(a, bfr, zacc[sub]);
        }
      }
    }
  }

  // Finalize: divide by row sums, store to z workspace [4096][1024]
  const float linv = 1.0f / lrow;
  float lj[8];
#pragma unroll
  for (int j = 0; j < 8; ++j)
    lj[j] = __shfl(linv, (lane < 16) ? j : 24 + j, 32);
  const int b = bh >> 4, h = bh & 15;
#pragma unroll
  for (int sub = 0; sub < 4; ++sub) {
    const int col = h * HDIM + sub * 16 + ln;
#pragma unroll
    for (int j = 0; j < 8; ++j) {
      const int m = j + 8 * kl;
      Z[(size_t)(b * T_SEQ + qbase + m) * D_MOD + col] = zacc[sub][j] * lj[j];
    }
  }
}

// ---------------------------------------------------------------------------
extern "C" void kernel_launch(void* const* d_in, const int* in_sizes, int n_in,
                              void* d_out, int out_size, void* d_ws, size_t ws_size,
                              hipStream_t stream) {
  const float* x = (const float*)d_in[0];      // [2,2048,1024]
  const float* W_attn = (const float*)d_in[1]; // [3072,1024]
  const float* W_out = (const float*)d_in[2];  // [1024,1024]
  float* out = (float*)d_out;                  // [2,2048,1024]
  float* ws = (float*)d_ws;

  float* q = ws;                        // QKV_PART floats each
  float* k = ws + (size_t)QKV_PART;
  float* v = ws + (size_t)2 * QKV_PART;
  float* z = ws + (size_t)3 * QKV_PART; // [4096][1024]

  dim3 blk(256);
  // 1) qkv = x @ W_attn^T, scattered to per-head layout
  gemm_xwt_kernel<<<dim3(32 * 48), blk, 0, stream>>>(x, W_attn, nullptr, ws,
                                                     3072, 48, 1);
  // 2) causal flash attention -> z
  attn_kernel<<<dim3(512), blk, 0, stream>>>(q, k, v, z);
  // 3) out = z @ W_out^T
  gemm_xwt_kernel<<<dim3(32 * 16), blk, 0, stream>>>(z, W_out, out, nullptr,
                                                     1024, 16, 0);
}